// BahdanauAttention_49074296324799
// MI455X (gfx1250) — compile-verified
//
#include <hip/hip_runtime.h>

#define BB 4
#define SS 512
#define HH 256

typedef float v2f __attribute__((ext_vector_type(2)));
typedef float v8f __attribute__((ext_vector_type(8)));

#if __has_builtin(__builtin_amdgcn_tanhf)
#define TANHF(x) __builtin_amdgcn_tanhf(x)
#else
#define TANHF(x) tanhf(x)
#endif

__device__ __forceinline__ v8f wmma_f32_16x16x4(v2f a, v2f b, v8f c) {
    // 8 args: (neg_a, A, neg_b, B, c_mod, C, reuse_a, reuse_b)
    return __builtin_amdgcn_wmma_f32_16x16x4_f32(false, a, false, b, (short)0, c,
                                                 false, false);
}

// ---------------------------------------------------------------------------
// Kernel A: Q = X @ W^T and K = X @ U^T.
// M = B*S = 2048 rows, N = K = H = 256. 16x16 tile per wave, 64 WMMA per tile.
// tiles 0..2047 -> Q/W ; tiles 2048..4095 -> K/U.
// ---------------------------------------------------------------------------
__global__ void bahdanau_proj_gemm(const float* __restrict__ X,
                                   const float* __restrict__ W,
                                   const float* __restrict__ U,
                                   float* __restrict__ Qo,
                                   float* __restrict__ Ko) {
    const int tid  = threadIdx.x;
    const int wave = tid >> 5;
    const int lane = tid & 31;
    const int hp   = lane >> 4;   // which half of the wave
    const int lid  = lane & 15;

    const int tile = blockIdx.x * 8 + wave;          // 0..4095
    const float* Bm = (tile & 2048) ? U : W;
    float*      Out = (tile & 2048) ? Ko : Qo;
    const int rem = tile & 2047;
    const int m0  = (rem >> 4) << 4;                 // 0..2032
    const int n0  = (rem & 15) << 4;                 // 0..240

    const float* arow = X  + (size_t)(m0 + lid) * HH;
    const float* brow = Bm + (size_t)(n0 + lid) * HH;

    v8f acc = {};
#pragma unroll 4
    for (int k = 0; k < HH; k += 4) {
        // A: 16x4 f32 -> lanes 0-15: K=k,k+1 ; lanes 16-31: K=k+2,k+3
        v2f a = *(const v2f*)(arow + k + 2 * hp);
        // B: 4x16 (B[k][n] = Wm[n][k]) -> same K split across halves
        v2f b = *(const v2f*)(brow + k + 2 * hp);
        acc = wmma_f32_16x16x4(a, b, acc);
    }

    float* orow = Out + n0 + lid;
#pragma unroll
    for (int r = 0; r < 8; ++r)
        orow[(size_t)(m0 + r + 8 * hp) * HH] = acc[r];
}

// ---------------------------------------------------------------------------
// Kernel B: per (b, 16-query tile):
//   phase 1: scores[16][512] via hardware v_tanh_f32
//   phase 2: row softmax in LDS
//   phase 3: context = weights @ X[b] via WMMA f32 16x16x4
// ---------------------------------------------------------------------------
__global__ void __launch_bounds__(256)
bahdanau_attn(const float* __restrict__ X,
              const float* __restrict__ Qb,
              const float* __restrict__ Kb,
              const float* __restrict__ v,
              float* __restrict__ out) {
    __shared__ float Qs[16 * HH];        // 16 KB: query tile
    __shared__ float sc[16 * SS];        // 32 KB: scores -> weights
    __shared__ float vs[HH];             // 1 KB

    const int tid = threadIdx.x;
    const int b   = blockIdx.x >> 5;
    const int q0  = (blockIdx.x & 31) << 4;

    // stage query tile + v into LDS (float4, fully coalesced)
    {
        const float4* src = (const float4*)(Qb + ((size_t)b * SS + q0) * HH);
        float4*       dst = (float4*)Qs;
        for (int i = tid; i < (16 * HH) / 4; i += 256) dst[i] = src[i];
        vs[tid] = v[tid];                // tid < 256 == HH
    }
    __syncthreads();

    // ---- phase 1: scores --------------------------------------------------
    {
        const int q  = tid >> 4;
        const int kk = tid & 15;
        const float* qrow = Qs + q * HH;
        for (int kb = 0; kb < SS; kb += 16) {
            const float* krow = Kb + ((size_t)b * SS + kb + kk) * HH;
            float acc = 0.f;
#pragma unroll 4
            for (int h = 0; h < HH; h += 4) {
                float4 qv = *(const float4*)(qrow + h);
                float4 kv = *(const float4*)(krow + h);
                float4 vv = *(const float4*)(vs + h);
                acc += vv.x * TANHF(qv.x + kv.x);
                acc += vv.y * TANHF(qv.y + kv.y);
                acc += vv.z * TANHF(qv.z + kv.z);
                acc += vv.w * TANHF(qv.w + kv.w);
            }
            sc[q * SS + kb + kk] = acc;
        }
    }
    __syncthreads();

    // ---- phase 2: softmax over k (wave handles 2 rows) --------------------
    {
        const int wv = tid >> 5;
        const int ln = tid & 31;
        for (int r = 0; r < 2; ++r) {
            float* row = sc + (wv * 2 + r) * SS;
            float m = -__builtin_inff();
            for (int k = ln; k < SS; k += 32) m = fmaxf(m, row[k]);
            for (int off = 16; off > 0; off >>= 1)
                m = fmaxf(m, __shfl_xor(m, off, 32));
            float s = 0.f;
            for (int k = ln; k < SS; k += 32) {
                float e = __builtin_amdgcn_exp2f((row[k] - m) * 1.4426950408889634f);
                row[k] = e;
                s += e;
            }
            for (int off = 16; off > 0; off >>= 1) s += __shfl_xor(s, off, 32);
            const float inv = 1.0f / s;
            for (int k = ln; k < SS; k += 32) row[k] *= inv;
        }
    }
    __syncthreads();

    // ---- phase 3: context = weights(16x512) @ X[b](512x256) via WMMA ------
    {
        const int wv  = tid >> 5;
        const int ln  = tid & 31;
        const int hp  = ln >> 4;
        const int lid = ln & 15;
        const float* xb = X + (size_t)b * SS * HH;

        for (int t = 0; t < 2; ++t) {
            const int n0 = wv * 32 + t * 16;    // 8 waves x 2 tiles = 256 cols
            v8f acc = {};
#pragma unroll 2
            for (int k = 0; k < SS; k += 4) {
                v2f a = *(const v2f*)(sc + lid * SS + k + 2 * hp);   // LDS b64
                v2f bf;
                bf.x = xb[(size_t)(k + 2 * hp)     * HH + n0 + lid];
                bf.y = xb[(size_t)(k + 2 * hp + 1) * HH + n0 + lid];
                acc = wmma_f32_16x16x4(a, bf, acc);
            }
#pragma unroll
            for (int r = 0; r < 8; ++r)
                out[((size_t)b * SS + q0 + r + 8 * hp) * HH + n0 + lid] = acc[r];
        }
    }
}

// ---------------------------------------------------------------------------
extern "C" void kernel_launch(void* const* d_in, const int* in_sizes, int n_in,
                              void* d_out, int out_size, void* d_ws, size_t ws_size,
                              hipStream_t stream) {
    const float* X = (const float*)d_in[0];   // (B,S,H)
    const float* W = (const float*)d_in[1];   // (H,H)
    const float* U = (const float*)d_in[2];   // (H,H)
    const float* v = (const float*)d_in[3];   // (H)
    float* out = (float*)d_out;               // (B,S,H)

    float* Qb = (float*)d_ws;                       // 2 MB
    float* Kb = Qb + (size_t)BB * SS * HH;          // 2 MB

    // 4096 16x16 tiles (Q and K GEMMs), 8 waves per 256-thread block
    bahdanau_proj_gemm<<<512, 256, 0, stream>>>(X, W, U, Qb, Kb);
    // one block per (b, 16-query tile)
    bahdanau_attn<<<BB * (SS / 16), 256, 0, stream>>>(X, Qb, Kb, v, out);
}